// RFGCN_42511586296563
// MI455X (gfx1250) — compile-verified
//
#include <hip/hip_runtime.h>
#include <stdint.h>

// ---------------------------------------------------------------------------
// RF-GCN pipeline for MI455X (gfx1250, wave32).
// Dense GEMMs: V_WMMA_F32_16X16X4_F32 (exact f32 matrix path on CDNA5), with
// the B panel staged into LDS by the Tensor Data Mover (tensor_load_to_lds,
// TENSORcnt) and shared by all 8 waves of the block. The K-loop is software-
// pipelined (double-buffered A/B fragments) so loads for chunk k+16 overlap
// the 4 WMMAs of chunk k. A rows are prefetched with global_prefetch.
// Scatter aggregation uses global f32 atomics.
// ---------------------------------------------------------------------------

#define KTOP 5

typedef __attribute__((ext_vector_type(2))) float v2f;
typedef __attribute__((ext_vector_type(8))) float v8f;
typedef __attribute__((ext_vector_type(4))) unsigned int v4u;
typedef __attribute__((ext_vector_type(8))) int v8i;
typedef __attribute__((ext_vector_type(4))) int v4i;

__device__ __forceinline__ float wred_max(float v) {
#pragma unroll
  for (int o = 16; o; o >>= 1) v = fmaxf(v, __shfl_xor(v, o, 32));
  return v;
}
__device__ __forceinline__ float wred_sum(float v) {
#pragma unroll
  for (int o = 16; o; o >>= 1) v += __shfl_xor(v, o, 32);
  return v;
}
__device__ __forceinline__ float leaky02(float x) { return x > 0.f ? x : 0.2f * x; }
__device__ __forceinline__ float sigmoidf(float x) { return 1.f / (1.f + __expf(-x)); }

__device__ __forceinline__ void atomicMaxF(float* a, float v) {
  int old = __float_as_int(*a);
  while (__int_as_float(old) < v) {
    int assumed = old;
    old = atomicCAS((int*)a, assumed, __float_as_int(v));
    if (old == assumed) break;
  }
}

// edge e in [0, NK): src = e/KTOP, dst = col[e]; e in [NK, NK+N): self loop.
__device__ __forceinline__ void edge_decode(int e, int NK, const int* __restrict__ col,
                                            int& src, int& dst) {
  if (e < NK) { src = e / KTOP; dst = col[e]; }
  else        { src = dst = e - NK; }
}

// ---------------------------------------------------------------------------
// TDM: stage a 2D panel of f32 into LDS (offset 0 of the kernel's static LDS).
// D# per CDNA5 ISA ch.8: group0 {count=1, lds_addr, global_addr[56:0], type=2},
// group1 {data_size=4B, tensor_dim0/1, tile_dim0/1, tensor_dim0_stride}.
// ---------------------------------------------------------------------------
__device__ __forceinline__ void tdm_load_panel_f32(const float* gptr,
                                                   uint32_t tdim0, uint32_t tdim1,
                                                   uint32_t tile0, uint32_t tile1,
                                                   uint64_t stride0_elems) {
  uint64_t ga = (uint64_t)(uintptr_t)gptr;
  v4u g0;
  g0.x = 1u;                                               // count=1 (valid), user mode
  g0.y = 0u;                                               // lds_addr = base of static LDS
  g0.z = (uint32_t)(ga & 0xFFFFFFFFu);                     // global_addr[31:0]
  g0.w = (uint32_t)((ga >> 32) & 0x01FFFFFFu) | (2u << 30);// global_addr[56:32], type=2
  v8i g1;
  g1[0] = (int)(2u << 16);                                 // data_size=2 -> 4B; wg_mask=0
  g1[1] = (int)((tdim0 & 0xFFFFu) << 16);                  // tensor_dim0[15:0] @63:48
  g1[2] = (int)((tdim0 >> 16) | ((tdim1 & 0xFFFFu) << 16));// tdim0[31:16], tdim1[15:0]
  g1[3] = (int)((tdim1 >> 16) | (tile0 << 16));            // tdim1[31:16], tile_dim0
  g1[4] = (int)(tile1 & 0xFFFFu);                          // tile_dim1; tile_dim2=0
  g1[5] = (int)(stride0_elems & 0xFFFFFFFFu);              // dim0 stride[31:0]
  g1[6] = (int)((stride0_elems >> 32) & 0xFFFFu);          // stride[47:32]; stride1=0
  g1[7] = 0;
  v4i z4 = {0, 0, 0, 0};
#if defined(__clang_major__) && __clang_major__ >= 23
  v8i z8 = {0, 0, 0, 0, 0, 0, 0, 0};
  __builtin_amdgcn_tensor_load_to_lds(g0, g1, z4, z4, z8, 0);
#else
  __builtin_amdgcn_tensor_load_to_lds(g0, g1, z4, z4, 0);
#endif
  __builtin_amdgcn_s_wait_tensorcnt(0);
}

// Fetch the 4 A-fragments + 4 B-fragments of one 16-K chunk into registers.
template <int TB>
__device__ __forceinline__ void load_frags(const float* __restrict__ arow,
                                           const float* __restrict__ ldsB,
                                           int kdim, int half, int l15, int kbase,
                                           v2f a[4], v2f b[4]) {
#pragma unroll
  for (int u = 0; u < 4; ++u) {
    const int ks = kbase + 4 * u + (half << 1);
    a[u].x = arow[ks];
    a[u].y = arow[ks + 1];
    if (TB) {
      const float* bp = &ldsB[(size_t)l15 * kdim + ks];
      b[u].x = bp[0]; b[u].y = bp[1];
    } else {
      b[u].x = ldsB[ks * 16 + l15];
      b[u].y = ldsB[(ks + 1) * 16 + l15];
    }
  }
}

// ---------------------------------------------------------------------------
// WMMA f32 GEMM: C[rows,cols] = A[rows,kdim] @ B (+bias)(+relu)
// Block = 8 waves = 128 rows x 16 cols. The kdim x 16 B panel is TDM-loaded
// into LDS once per block; each wave runs a branch-free, double-buffered
// K-loop: 4x v_wmma_f32_16x16x4_f32 per chunk while next chunk's fragments
// load. rows%16==0, cols%16==0, kdim%16==0 at every call site.
// ---------------------------------------------------------------------------
template <int TB>
__global__ void gemm_wmma_kernel(const float* __restrict__ A, int lda,
                                 const float* __restrict__ B, int ldb,
                                 const float* __restrict__ bias,
                                 float* __restrict__ C, int ldc,
                                 int rows, int kdim, int cols, int act) {
  __shared__ float ldsB[4096];                      // 16 KB: up to 256-K x 16-col panel
  const int lane = threadIdx.x & 31;
  const int wave = threadIdx.x >> 5;
  const int tiles_n = cols >> 4;
  const int tn = (blockIdx.x % tiles_n) << 4;
  const int tm = (blockIdx.x / tiles_n) * 128 + (wave << 4);
  const int half = lane >> 4;                       // 0: K0/K1 ; 1: K2/K3
  const int l15 = lane & 15;

  if (wave == 0) {                                  // one TDM op per block
    if (TB) tdm_load_panel_f32(B + (size_t)tn * ldb, (uint32_t)kdim, 16u,
                               (uint32_t)kdim, 16u, (uint64_t)ldb);
    else    tdm_load_panel_f32(B + tn, (uint32_t)cols, (uint32_t)kdim,
                               16u, (uint32_t)kdim, (uint64_t)ldb);
  }
  __syncthreads();                                  // panel visible to all waves

  if (tm < rows) {                                  // wave-uniform: EXEC all-1s at WMMA
    const float* arow = A + (size_t)(tm + l15) * lda;
    const int nn = tn + l15;
    v8f acc = {};
    v2f a0[4], b0[4], a1[4], b1[4];
    load_frags<TB>(arow, ldsB, kdim, half, l15, 0, a0, b0);
    int k16 = 0;
    for (; k16 + 16 < kdim; k16 += 16) {            // steady state: prefetch + overlap
      __builtin_prefetch(arow + k16 + 32, 0, 3);    // global_prefetch 2 chunks ahead
      load_frags<TB>(arow, ldsB, kdim, half, l15, k16 + 16, a1, b1);
#pragma unroll
      for (int u = 0; u < 4; ++u)
        acc = __builtin_amdgcn_wmma_f32_16x16x4_f32(false, a0[u], false, b0[u],
                                                    (short)0, acc, false, false);
#pragma unroll
      for (int u = 0; u < 4; ++u) { a0[u] = a1[u]; b0[u] = b1[u]; }
    }
#pragma unroll
    for (int u = 0; u < 4; ++u)                     // peeled final chunk
      acc = __builtin_amdgcn_wmma_f32_16x16x4_f32(false, a0[u], false, b0[u],
                                                  (short)0, acc, false, false);
    const float badd = bias ? bias[nn] : 0.f;
#pragma unroll
    for (int g = 0; g < 8; ++g) {
      const int r = tm + g + (half << 3);
      float v = acc[g] + badd;
      if (act == 1) v = fmaxf(v, 0.f);
      C[(size_t)r * ldc + nn] = v;
    }
  }
}

// ---------------------------------------------------------------------------
// Attention edge build: per row n, 4-head softmax over all m, head-mean, top-5.
// One wave per row, three streaming passes over k (L2-resident).
// ---------------------------------------------------------------------------
__device__ __forceinline__ void head_scores(const float* __restrict__ qv,
                                            const float* __restrict__ kr, float* s) {
#pragma unroll
  for (int h = 0; h < 4; ++h)
    s[h] = qv[4*h+0]*kr[4*h+0] + qv[4*h+1]*kr[4*h+1] +
           qv[4*h+2]*kr[4*h+2] + qv[4*h+3]*kr[4*h+3];
}

__global__ void attn_topk_kernel(const float* __restrict__ qkv, int* __restrict__ col,
                                 int n_nodes) {
  const int n = blockIdx.x;
  const int lane = threadIdx.x;
  float qv[16];
  const float* qrow = qkv + (size_t)n * 48;
#pragma unroll
  for (int i = 0; i < 16; ++i) qv[i] = qrow[i] * 0.5f;   // fold 1/sqrt(HD)

  float mx[4] = {-1e30f, -1e30f, -1e30f, -1e30f};
  for (int m = lane; m < n_nodes; m += 32) {
    const float* kr = qkv + (size_t)m * 48 + 16;
    float s[4]; head_scores(qv, kr, s);
#pragma unroll
    for (int h = 0; h < 4; ++h) mx[h] = fmaxf(mx[h], s[h]);
  }
#pragma unroll
  for (int h = 0; h < 4; ++h) mx[h] = wred_max(mx[h]);

  float sm[4] = {0.f, 0.f, 0.f, 0.f};
  for (int m = lane; m < n_nodes; m += 32) {
    const float* kr = qkv + (size_t)m * 48 + 16;
    float s[4]; head_scores(qv, kr, s);
#pragma unroll
    for (int h = 0; h < 4; ++h) sm[h] += __expf(s[h] - mx[h]);
  }
  float inv[4];
#pragma unroll
  for (int h = 0; h < 4; ++h) inv[h] = 0.25f / wred_sum(sm[h]);

  float bv[KTOP]; int bi[KTOP];
#pragma unroll
  for (int j = 0; j < KTOP; ++j) { bv[j] = -1e30f; bi[j] = 0; }
  for (int m = lane; m < n_nodes; m += 32) {
    const float* kr = qkv + (size_t)m * 48 + 16;
    float s[4]; head_scores(qv, kr, s);
    float a = 0.f;
#pragma unroll
    for (int h = 0; h < 4; ++h) a += __expf(s[h] - mx[h]) * inv[h];
    if (a > bv[KTOP - 1]) {
      int j = KTOP - 1;
      while (j > 0 && a > bv[j - 1]) { bv[j] = bv[j-1]; bi[j] = bi[j-1]; --j; }
      bv[j] = a; bi[j] = m;
    }
  }
  __shared__ float sv[32 * KTOP];
  __shared__ int   si[32 * KTOP];
#pragma unroll
  for (int j = 0; j < KTOP; ++j) { sv[lane * KTOP + j] = bv[j]; si[lane * KTOP + j] = bi[j]; }
  __syncthreads();
  if (lane == 0) {
    for (int j = 0; j < KTOP; ++j) {
      float best = -1e30f; int besti = 0, bestp = 0;
      for (int t = 0; t < 32 * KTOP; ++t)
        if (sv[t] > best) { best = sv[t]; besti = si[t]; bestp = t; }
      sv[bestp] = -1e30f;
      col[(size_t)n * KTOP + j] = besti;
    }
  }
}

// ---------------------------------------------------------------------------
// GAT edge pipeline: per-node attention dots, segment max/sum, scatter agg.
// ---------------------------------------------------------------------------
__global__ void gat_rowdot_kernel(const float* __restrict__ hW,
                                  const float* __restrict__ asrc,
                                  const float* __restrict__ adst,
                                  float* a_s, float* a_d,
                                  float* segmax, float* segsum, int ccols) {
  const int n = blockIdx.x, lane = threadIdx.x;
  const float* xr = hW + (size_t)n * ccols;
  float s1 = 0.f, s2 = 0.f;
  for (int c = lane; c < ccols; c += 32) { float v = xr[c]; s1 += v * asrc[c]; s2 += v * adst[c]; }
  s1 = wred_sum(s1); s2 = wred_sum(s2);
  if (!lane) { a_s[n] = s1; a_d[n] = s2; segmax[n] = -1e30f; segsum[n] = 0.f; }
}

__global__ void edge_max_kernel(const float* __restrict__ a_s, const float* __restrict__ a_d,
                                const int* __restrict__ col, float* segmax, int NK, int NE) {
  int e = blockIdx.x * blockDim.x + threadIdx.x;
  if (e >= NE) return;
  int src, dst; edge_decode(e, NK, col, src, dst);
  atomicMaxF(&segmax[dst], leaky02(a_s[src] + a_d[dst]));
}

__global__ void edge_sum_kernel(const float* __restrict__ a_s, const float* __restrict__ a_d,
                                const int* __restrict__ col, const float* __restrict__ segmax,
                                float* segsum, int NK, int NE) {
  int e = blockIdx.x * blockDim.x + threadIdx.x;
  if (e >= NE) return;
  int src, dst; edge_decode(e, NK, col, src, dst);
  atomicAdd(&segsum[dst], __expf(leaky02(a_s[src] + a_d[dst]) - segmax[dst]));
}

__global__ void edge_gat_agg_kernel(const float* __restrict__ a_s, const float* __restrict__ a_d,
                                    const int* __restrict__ col, const float* __restrict__ segmax,
                                    const float* __restrict__ segsum, const float* __restrict__ hW,
                                    float* __restrict__ out, int NK, int dout) {
  const int e = blockIdx.x, c = threadIdx.x;
  int src, dst; edge_decode(e, NK, col, src, dst);
  float alpha = __expf(leaky02(a_s[src] + a_d[dst]) - segmax[dst]) / segsum[dst];
  atomicAdd(&out[(size_t)dst * dout + c], alpha * hW[(size_t)src * dout + c]);
}

// ---------------------------------------------------------------------------
// GCN: degree count, rsqrt, normalized scatter agg.
// ---------------------------------------------------------------------------
__global__ void deg_kernel(const int* __restrict__ col, float* deg, int NK, int NE) {
  int e = blockIdx.x * blockDim.x + threadIdx.x;
  if (e >= NE) return;
  int src, dst; edge_decode(e, NK, col, src, dst);
  (void)src;
  atomicAdd(&deg[dst], 1.f);
}
__global__ void dinv_kernel(float* deg, int n) {
  int i = blockIdx.x * blockDim.x + threadIdx.x;
  if (i < n) deg[i] = rsqrtf(fmaxf(deg[i], 1.f));
}
__global__ void edge_gcn_agg_kernel(const int* __restrict__ col, const float* __restrict__ dinv,
                                    const float* __restrict__ hW, float* __restrict__ out,
                                    int NK, int dout) {
  const int e = blockIdx.x, c = threadIdx.x;
  int src, dst; edge_decode(e, NK, col, src, dst);
  float nrm = dinv[src] * dinv[dst];
  atomicAdd(&out[(size_t)dst * dout + c], nrm * hW[(size_t)src * dout + c]);
}

// ---------------------------------------------------------------------------
// BatchNorm (training-mode batch stats over nodes), LayerNorm, misc.
// ---------------------------------------------------------------------------
__global__ void bn_stats_kernel(const float* __restrict__ x, float* mean, float* varr,
                                int n, int c) {
  __shared__ float s1[256], s2[256];
  const int ch = blockIdx.x, tid = threadIdx.x;
  float a = 0.f, b = 0.f;
  for (int i = tid; i < n; i += 256) { float v = x[(size_t)i * c + ch]; a += v; b += v * v; }
  s1[tid] = a; s2[tid] = b; __syncthreads();
  for (int o = 128; o; o >>= 1) {
    if (tid < o) { s1[tid] += s1[tid + o]; s2[tid] += s2[tid + o]; }
    __syncthreads();
  }
  if (!tid) { float m = s1[0] / n; mean[ch] = m; varr[ch] = fmaxf(s2[0] / n - m * m, 0.f); }
}

__global__ void bn_apply_kernel(const float* __restrict__ x, const float* __restrict__ mean,
                                const float* __restrict__ varr, const float* __restrict__ g,
                                const float* __restrict__ b, const float* __restrict__ resid,
                                float* __restrict__ y, int total, int c, int act) {
  int i = blockIdx.x * blockDim.x + threadIdx.x;
  if (i >= total) return;
  int ch = i % c;
  float v = (x[i] - mean[ch]) * rsqrtf(varr[ch] + 1e-5f) * g[ch] + b[ch];
  if (resid) v += resid[i];
  if (act == 1) v = fmaxf(v, 0.f);
  y[i] = v;
}

__global__ void layernorm_kernel(const float* __restrict__ x, const float* __restrict__ g,
                                 const float* __restrict__ b, float* __restrict__ y,
                                 int c, int act) {
  const int n = blockIdx.x, lane = threadIdx.x;   // one wave per row
  const float* xr = x + (size_t)n * c;
  float s = 0.f, s2 = 0.f;
  for (int i = lane; i < c; i += 32) { float v = xr[i]; s += v; s2 += v * v; }
  s = wred_sum(s); s2 = wred_sum(s2);
  float m = s / c, var = fmaxf(s2 / c - m * m, 0.f);
  float inv = rsqrtf(var + 1e-5f);
  for (int i = lane; i < c; i += 32) {
    float v = (xr[i] - m) * inv * g[i] + b[i];
    if (act == 1) v = fmaxf(v, 0.f);
    else if (act == 2) v = sigmoidf(v);
    y[(size_t)n * c + i] = v;
  }
}

__global__ void rowdot_kernel(const float* __restrict__ x, const float* __restrict__ w,
                              const float* __restrict__ bias, float* __restrict__ y,
                              int c, int ystride, int act) {
  const int n = blockIdx.x, lane = threadIdx.x;
  const float* xr = x + (size_t)n * c;
  float s = 0.f;
  for (int i = lane; i < c; i += 32) s += xr[i] * w[i];
  s = wred_sum(s);
  if (!lane) {
    float v = s + bias[0];
    if (act == 2) v = sigmoidf(v);
    y[(size_t)n * ystride] = v;
  }
}

__global__ void node_softmax_stats_kernel(const float* __restrict__ logit, float* stats, int n) {
  __shared__ float red[1024];
  const int tid = threadIdx.x;
  float m = -1e30f;
  for (int i = tid; i < n; i += 1024) m = fmaxf(m, logit[i]);
  red[tid] = m; __syncthreads();
  for (int o = 512; o; o >>= 1) { if (tid < o) red[tid] = fmaxf(red[tid], red[tid + o]); __syncthreads(); }
  const float mx = red[0]; __syncthreads();
  float s = 0.f;
  for (int i = tid; i < n; i += 1024) s += __expf(logit[i] - mx);
  red[tid] = s; __syncthreads();
  for (int o = 512; o; o >>= 1) { if (tid < o) red[tid] += red[tid + o]; __syncthreads(); }
  if (!tid) { stats[0] = mx; stats[1] = red[0]; }
}

__global__ void node_softmax_apply_kernel(float* __restrict__ r, const float* __restrict__ logit,
                                          const float* __restrict__ stats, int total, int c) {
  int i = blockIdx.x * blockDim.x + threadIdx.x;
  if (i >= total) return;
  r[i] *= __expf(logit[i / c] - stats[0]) / stats[1];
}

// comb[:, :128] = c*cw*1.1 ; comb[:, 128:] = c*gate[n]*0.9
__global__ void cqi_comb_kernel(const float* __restrict__ cb, const float* __restrict__ cw,
                                const float* __restrict__ gate, float* __restrict__ comb,
                                int total) {
  int i = blockIdx.x * blockDim.x + threadIdx.x;
  if (i >= total) return;
  int n = i / 128, ch = i % 128;
  comb[(size_t)n * 256 + ch]       = cb[i] * cw[i] * 1.1f;
  comb[(size_t)n * 256 + 128 + ch] = cb[i] * gate[n] * 0.9f;
}

__global__ void add_kernel(const float* __restrict__ a, const float* __restrict__ b,
                           float* __restrict__ y, int total) {
  int i = blockIdx.x * blockDim.x + threadIdx.x;
  if (i < total) y[i] = a[i] + b[i];
}

// ---------------------------------------------------------------------------
// Host orchestration.
// Input index map (setup_inputs insertion order, depth-first):
//  0:x  1:fp.w 2:fp.b 3:attn_in_w 4:attn_in_b
//  5-8:conv1{w,b,a_src,a_dst} 9-12:conv2 13-16:conv3
//  17,18:rssi_conv1{w,b} 19,20:rssi_conv2 21,22:rssi_att
//  23-32:rssi_pred[lin,ln,lin,ln,lin] 33,34:cqi_conv1 35,36:cqi_conv2
//  37-44:cqi_ch 45-54:cqi_sp 55-62:cqi_fuse 63-72:cqi_pred
//  73-78:bn1..bn3 79,80:bn_rssi1 81,82:bn_rssi2 83,84:bn_cqi1 85,86:bn_cqi2
// ---------------------------------------------------------------------------
extern "C" void kernel_launch(void* const* d_in, const int* in_sizes, int n_in,
                              void* d_out, int out_size, void* d_ws, size_t ws_size,
                              hipStream_t stream) {
  (void)n_in; (void)out_size; (void)ws_size;
  const int N = in_sizes[0] / 32;           // F_IN = 32
  const int NK = N * KTOP;
  const int NE = NK + N;
  const float* x = (const float*)d_in[0];
  auto P = [&](int i) { return (const float*)d_in[i]; };

  float* W = (float*)d_ws;
  size_t off = 0;
  auto alloc = [&](size_t n) { float* p = W + off; off += n; return p; };
  float* xp     = alloc((size_t)N * 16);
  float* qkv    = alloc((size_t)N * 48);
  float* a_s    = alloc(N);
  float* a_d    = alloc(N);
  float* segmax = alloc(N);
  float* segsum = alloc(N);
  float* dinv   = alloc(N);      // degree then rsqrt(degree)
  float* logit  = alloc(N);
  float* stats  = alloc(8);
  float* mean   = alloc(256);
  float* varr   = alloc(256);
  int*   col    = (int*)alloc(NK);
  float* bufA   = alloc((size_t)N * 256);
  float* bufB   = alloc((size_t)N * 256);
  float* bufC   = alloc((size_t)N * 256);
  float* bufD   = alloc((size_t)N * 256);

  // rows are always N (multiple of 128); block = 8 waves = 128 rows x 16 cols.
  auto gemm = [&](const float* A, int lda, const float* B, int ldb, int tB,
                  const float* bias, float* C, int ldc, int rows, int kk, int cols, int act) {
    int blocks = ((rows + 127) / 128) * (cols / 16);
    if (tB) gemm_wmma_kernel<1><<<blocks, 256, 0, stream>>>(A, lda, B, ldb, bias, C, ldc,
                                                            rows, kk, cols, act);
    else    gemm_wmma_kernel<0><<<blocks, 256, 0, stream>>>(A, lda, B, ldb, bias, C, ldc,
                                                            rows, kk, cols, act);
  };
  const int EB = (NE + 255) / 256;

  auto gat_layer = [&](const float* xin, int din, int pi, const float* bng, const float* bnb,
                       const float* resid, float* hW, float* agg, float* out) {
    gemm(xin, din, P(pi), 256, 0, nullptr, hW, 256, N, din, 256, 0);   // bias cancels in BN
    gat_rowdot_kernel<<<N, 32, 0, stream>>>(hW, P(pi + 2), P(pi + 3), a_s, a_d, segmax, segsum, 256);
    edge_max_kernel<<<EB, 256, 0, stream>>>(a_s, a_d, col, segmax, NK, NE);
    edge_sum_kernel<<<EB, 256, 0, stream>>>(a_s, a_d, col, segmax, segsum, NK, NE);
    hipMemsetAsync(agg, 0, (size_t)N * 256 * sizeof(float), stream);
    edge_gat_agg_kernel<<<NE, 256, 0, stream>>>(a_s, a_d, col, segmax, segsum, hW, agg, NK, 256);
    bn_stats_kernel<<<256, 256, 0, stream>>>(agg, mean, varr, N, 256);
    bn_apply_kernel<<<(N * 256 + 255) / 256, 256, 0, stream>>>(agg, mean, varr, bng, bnb,
                                                               resid, out, N * 256, 256, 1);
  };
  auto gcn_layer = [&](const float* xin, int din, const float* w, int dout,
                       const float* bng, const float* bnb, float* hW, float* agg, float* out) {
    gemm(xin, din, w, dout, 0, nullptr, hW, dout, N, din, dout, 0);    // bias cancels in BN
    hipMemsetAsync(agg, 0, (size_t)N * dout * sizeof(float), stream);
    edge_gcn_agg_kernel<<<NE, dout, 0, stream>>>(col, dinv, hW, agg, NK, dout);
    bn_stats_kernel<<<dout, 256, 0, stream>>>(agg, mean, varr, N, dout);
    bn_apply_kernel<<<(N * dout + 255) / 256, 256, 0, stream>>>(agg, mean, varr, bng, bnb,
                                                                nullptr, out, N * dout, dout, 1);
  };

  // ---- Stage 1: edge construction --------------------------------------
  gemm(x, 32, P(1), 16, 0, P(2), xp, 16, N, 32, 16, 0);                 // feature_proj
  gemm(xp, 16, P(3), 16, 1, P(4), qkv, 48, N, 16, 48, 0);               // qkv = xp @ W^T + b
  attn_topk_kernel<<<N, 32, 0, stream>>>(qkv, col, N);
  hipMemsetAsync(dinv, 0, N * sizeof(float), stream);
  deg_kernel<<<EB, 256, 0, stream>>>(col, dinv, NK, NE);
  dinv_kernel<<<(N + 255) / 256, 256, 0, stream>>>(dinv, N);

  // ---- GAT stack --------------------------------------------------------
  gat_layer(x, 32, 5, P(73), P(74), nullptr, bufA, bufB, bufC);         // h1 -> bufC
  gat_layer(bufC, 256, 9, P(75), P(76), bufC, bufA, bufB, bufD);        // h2 -> bufD
  gat_layer(bufD, 256, 13, P(77), P(78), bufD, bufA, bufB, bufC);       // h  -> bufC
  float* h = bufC;

  // ---- RSSI branch ------------------------------------------------------
  gcn_layer(h, 256, P(17), 256, P(79), P(80), bufA, bufB, bufD);        // r1 -> bufD [N,256]
  gcn_layer(bufD, 256, P(19), 128, P(81), P(82), bufA, bufB, bufD);     // r  -> bufD [N,128]
  float* r = bufD;
  rowdot_kernel<<<N, 32, 0, stream>>>(r, P(21), P(22), logit, 128, 1, 0);
  node_softmax_stats_kernel<<<1, 1024, 0, stream>>>(logit, stats, N);
  node_softmax_apply_kernel<<<(N * 128 + 255) / 256, 256, 0, stream>>>(r, logit, stats, N * 128, 128);
  gemm(r, 128, P(23), 128, 0, P(24), bufA, 128, N, 128, 128, 0);
  layernorm_kernel<<<N, 32, 0, stream>>>(bufA, P(25), P(26), bufB, 128, 1);
  gemm(bufB, 128, P(27), 64, 0, P(28), bufA, 64, N, 128, 64, 0);
  layernorm_kernel<<<N, 32, 0, stream>>>(bufA, P(29), P(30), bufB, 64, 1);
  rowdot_kernel<<<N, 32, 0, stream>>>(bufB, P(31), P(32), (float*)d_out + 0, 64, 2, 0);

  // ---- CQI branch -------------------------------------------------------
  gcn_layer(h, 256, P(33), 256, P(83), P(84), bufA, bufB, bufD);        // c1 -> bufD [N,256]
  gcn_layer(bufD, 256, P(35), 128, P(85), P(86), bufA, bufB, bufD);     // c  -> bufD [N,128]
  float* c = bufD;                                                       // bufC (h) now free
  // channel gate: cw
  gemm(c, 128, P(37), 128, 0, P(38), bufA, 128, N, 128, 128, 0);
  layernorm_kernel<<<N, 32, 0, stream>>>(bufA, P(39), P(40), bufB, 128, 1);
  gemm(bufB, 128, P(41), 128, 0, P(42), bufA, 128, N, 128, 128, 0);
  layernorm_kernel<<<N, 32, 0, stream>>>(bufA, P(43), P(44), bufB, 128, 2);   // cw = bufB
  // spatial gate
  gemm(c, 128, P(45), 64, 0, P(46), bufA, 64, N, 128, 64, 0);
  layernorm_kernel<<<N, 32, 0, stream>>>(bufA, P(47), P(48), bufC, 64, 1);
  gemm(bufC, 64, P(49), 32, 0, P(50), bufA, 32, N, 64, 32, 0);
  layernorm_kernel<<<N, 32, 0, stream>>>(bufA, P(51), P(52), bufC, 32, 1);
  rowdot_kernel<<<N, 32, 0, stream>>>(bufC, P(53), P(54), logit, 32, 1, 2);   // gate (sigmoid)
  // combine + fuse + residual
  cqi_comb_kernel<<<(N * 128 + 255) / 256, 256, 0, stream>>>(c, bufB, logit, bufC, N * 128);
  gemm(bufC, 256, P(55), 128, 0, P(56), bufA, 128, N, 256, 128, 0);
  layernorm_kernel<<<N, 32, 0, stream>>>(bufA, P(57), P(58), bufB, 128, 1);
  gemm(bufB, 128, P(59), 128, 0, P(60), bufA, 128, N, 128, 128, 0);
  layernorm_kernel<<<N, 32, 0, stream>>>(bufA, P(61), P(62), bufB, 128, 1);
  add_kernel<<<(N * 128 + 255) / 256, 256, 0, stream>>>(bufB, c, bufC, N * 128);
  // cqi prediction head
  gemm(bufC, 128, P(63), 128, 0, P(64), bufA, 128, N, 128, 128, 0);
  layernorm_kernel<<<N, 32, 0, stream>>>(bufA, P(65), P(66), bufB, 128, 1);
  gemm(bufB, 128, P(67), 64, 0, P(68), bufA, 64, N, 128, 64, 0);
  layernorm_kernel<<<N, 32, 0, stream>>>(bufA, P(69), P(70), bufB, 64, 1);
  rowdot_kernel<<<N, 32, 0, stream>>>(bufB, P(71), P(72), (float*)d_out + 1, 64, 2, 0);
}